// R_transformations_84533546320185
// MI455X (gfx1250) — compile-verified
//
#include <hip/hip_runtime.h>
#include <math.h>

typedef __attribute__((ext_vector_type(2))) float v2f;
typedef __attribute__((ext_vector_type(8))) float v8f;

__device__ __forceinline__ float gelu_exact(float v) {
    // exact GELU: 0.5*x*(1+erf(x/sqrt(2)))
    return 0.5f * v * (1.0f + erff(v * 0.70710678118654752440f));
}

// One complex K-step (2 complex input channels across the lane halves):
//   A fragment (16x4 f32): lanes 0-15 hold K={4t,4t+1}=(ar,ai) of channel 2t,
//                          lanes 16-31 hold K={4t+2,4t+3} of channel 2t+1.
//   B fragment (4x16 f32): same K split; Bre=(wr,-wi), Bim=(wi,wr).
__device__ __forceinline__ void wmma_cstep(const float2 a0, const float2 a1,
                                           const float2 wv,
                                           v8f& cr0, v8f& ci0, v8f& cr1, v8f& ci1)
{
    const v2f A0  = { a0.x, a0.y };
    const v2f A1  = { a1.x, a1.y };
    const v2f Bre = { wv.x, -wv.y };
    const v2f Bim = { wv.y,  wv.x };
    cr0 = __builtin_amdgcn_wmma_f32_16x16x4_f32(false, A0, false, Bre, (short)0, cr0, false, false);
    ci0 = __builtin_amdgcn_wmma_f32_16x16x4_f32(false, A0, false, Bim, (short)0, ci0, false, false);
    cr1 = __builtin_amdgcn_wmma_f32_16x16x4_f32(false, A1, false, Bre, (short)0, cr1, false, false);
    ci1 = __builtin_amdgcn_wmma_f32_16x16x4_f32(false, A1, false, Bim, (short)0, ci1, false, false);
}

// dst = GELU( concat(srcA,srcB) x W + bias )   (complex, per spectral mode)
// One wave32 owns one (position, 16-wide c_out tile); M = batch = 32.
// CA/CB even; split K loops avoid any lane-divergent source select.
template<int CA, int SA, int CB, int SB, int COUT, int SLOC, int SDST, int ACC>
__global__ __launch_bounds__(256) void r_trans_wmma_kernel(
    const float2* __restrict__ srcA, int rowoffA, int strideA,
    const float2* __restrict__ srcB, int rowoffB, int strideB,
    const float2* __restrict__ w, const float2* __restrict__ bias,
    float2* __restrict__ dst, int rowoffD, int strideD,
    int m1, int m2, int n_tiles)
{
    const int lane  = threadIdx.x & 31;
    const int gw    = blockIdx.x * (blockDim.x >> 5) + (threadIdx.x >> 5);
    const int ntile = gw % n_tiles;
    const int pos   = gw / n_tiles;          // local spatial position of this level
    const int r     = pos / m2;
    const int c     = pos - r * m2;

    const int hi = lane >> 4;                // 0: even complex channel, 1: odd
    const int bb = lane & 15;
    const int o  = ntile * 16 + bb;          // output channel

    // crop remap into each source's own layout
    const int posA = (r + (r < m1 ? 0 : rowoffA)) * strideA + c;
    const int posB = (r + (r < m1 ? 0 : rowoffB)) * strideB + c;

    v8f cr0 = {}, ci0 = {}, cr1 = {}, ci1 = {};

    constexpr size_t WSTEP = (size_t)COUT * SLOC;    // elements per input channel of W
    const float2* pw  = w    + (size_t)hi * WSTEP + (size_t)o * SLOC + pos;
    const float2* pa0 = srcA + ((size_t)bb        * CA + hi) * SA + posA;  // batch 0..15
    const float2* pa1 = srcA + ((size_t)(bb + 16) * CA + hi) * SA + posA;  // batch 16..31

    #pragma unroll 4
    for (int t = 0; t < (CA >> 1); ++t) {
        const float2 a0 = *pa0;
        const float2 a1 = *pa1;
        const float2 wv = *pw;
        __builtin_prefetch(pw + 4 * WSTEP, 0, 3);    // global_prefetch_b8: weights 2 steps ahead
        pa0 += 2 * (size_t)SA;
        pa1 += 2 * (size_t)SA;
        pw  += 2 * WSTEP;
        wmma_cstep(a0, a1, wv, cr0, ci0, cr1, ci1);
    }

    if constexpr (CB > 0) {
        const float2* pb0 = srcB + ((size_t)bb        * CB + hi) * SB + posB;
        const float2* pb1 = srcB + ((size_t)(bb + 16) * CB + hi) * SB + posB;
        #pragma unroll 2
        for (int t = 0; t < (CB >> 1); ++t) {
            const float2 a0 = *pb0;
            const float2 a1 = *pb1;
            const float2 wv = *pw;
            pb0 += 2 * (size_t)SB;
            pb1 += 2 * (size_t)SB;
            pw  += 2 * WSTEP;
            wmma_cstep(a0, a1, wv, cr0, ci0, cr1, ci1);
        }
    }

    // bias depends only on (o, pos): every C element in this lane has N = o
    const float2 bv = bias[(size_t)o * SLOC + pos];
    const int posD = (r + (r < m1 ? 0 : rowoffD)) * strideD + c;  // corner remap for u-levels

    #pragma unroll
    for (int rr = 0; rr < 8; ++rr) {
        const int m0 = hi ? (8 + rr) : rr;   // C layout: lanes 0-15 M=rr, lanes 16-31 M=8+rr
        {
            float re = gelu_exact(cr0[rr] + bv.x);
            float im = gelu_exact(ci0[rr] + bv.y);
            const size_t oi = ((size_t)m0 * COUT + o) * SDST + posD;
            if constexpr (ACC) { float2 old = dst[oi]; re += old.x; im += old.y; }
            dst[oi] = make_float2(re, im);
        }
        {
            float re = gelu_exact(cr1[rr] + bv.x);
            float im = gelu_exact(ci1[rr] + bv.y);
            const size_t oi = ((size_t)(16 + m0) * COUT + o) * SDST + posD;
            if constexpr (ACC) { float2 old = dst[oi]; re += old.x; im += old.y; }
            dst[oi] = make_float2(re, im);
        }
    }
}

extern "C" void kernel_launch(void* const* d_in, const int* in_sizes, int n_in,
                              void* d_out, int out_size, void* d_ws, size_t ws_size,
                              hipStream_t stream)
{
    (void)in_sizes; (void)n_in; (void)out_size; (void)d_ws; (void)ws_size;
    const float2* x    = (const float2*)d_in[0];   // (32, 32, 128, 64) complex
    const float2* K    = (const float2*)d_in[1];   // (32, 12, 128, 64)
    const float2* w_d0 = (const float2*)d_in[2];
    const float2* b_d0 = (const float2*)d_in[3];
    const float2* w_d1 = (const float2*)d_in[4];
    const float2* b_d1 = (const float2*)d_in[5];
    const float2* w_u1 = (const float2*)d_in[6];
    const float2* b_u1 = (const float2*)d_in[7];
    const float2* w_d2 = (const float2*)d_in[8];
    const float2* b_d2 = (const float2*)d_in[9];
    const float2* w_u2 = (const float2*)d_in[10];
    const float2* b_u2 = (const float2*)d_in[11];

    float2* out = (float2*)d_out;
    float2* s0 = out;                        // d0  -> d0p       32*32*128*64
    float2* s1 = out + (size_t)8388608;      // duplicate of d0p
    float2* s2 = out + (size_t)16777216;     // d1  -> d1p       32*64*64*32
    float2* s3 = out + (size_t)20971520;     // d2               32*128*32*16

    const dim3 blk(256);  // 8 waves/block; every launch has wave-count % 8 == 0

    // d0 = r_trans([x ; K], w_d0)          S=8192 (128x64), c_in=44, c_out=32
    r_trans_wmma_kernel<32, 8192, 12, 8192, 32, 8192, 8192, 0><<<2048, blk, 0, stream>>>(
        x, 0, 64,   K, 0, 64,   w_d0, b_d0,  s0, 0, 64,  64, 64, 2);

    // d1 = r_trans([crop(d0) ; crop(K)])   S=2048 (64x32), c_in=44, c_out=64
    r_trans_wmma_kernel<32, 8192, 12, 8192, 64, 2048, 2048, 0><<<1024, blk, 0, stream>>>(
        s0, 64, 64,  K, 64, 64,  w_d1, b_d1,  s2, 0, 32,  32, 32, 4);

    // d2 = r_trans([crop(d1) ; crop(K)])   S=512 (32x16), c_in=76, c_out=128
    r_trans_wmma_kernel<64, 2048, 12, 8192, 128, 512, 512, 0><<<512, blk, 0, stream>>>(
        s2, 32, 32,  K, 96, 64,  w_d2, b_d2,  s3, 0, 16,  16, 16, 8);

    // u2 = r_trans(d2); d1p = corner_add(d1, u2)  -> accumulate into s2 corners
    r_trans_wmma_kernel<128, 512, 0, 1, 64, 512, 2048, 1><<<256, blk, 0, stream>>>(
        s3, 0, 16,   s3, 0, 1,   w_u2, b_u2,  s2, 32, 32, 16, 16, 4);

    // u1 = r_trans(d1p); d0p = corner_add(d0, u1) -> accumulate into s0 corners
    r_trans_wmma_kernel<64, 2048, 0, 1, 32, 2048, 8192, 1><<<512, blk, 0, stream>>>(
        s2, 0, 32,   s2, 0, 1,   w_u1, b_u1,  s0, 64, 64, 32, 32, 2);

    // outputs are (d0p, d0p, d1p, d2): duplicate slot0 into slot1
    hipMemcpyAsync(s1, s0, (size_t)8388608 * sizeof(float2),
                   hipMemcpyDeviceToDevice, stream);
}